// NER_87531433492674
// MI455X (gfx1250) — compile-verified
//
#include <hip/hip_runtime.h>
#include <hip/hip_bf16.h>
#include <math.h>

// Problem constants (from reference)
#define B_  8
#define S_  256
#define W_  5
#define E_  128
#define HD  256
#define F_  133   // E + W
#define T_  20

typedef float v2f __attribute__((ext_vector_type(2)));
typedef float v8f __attribute__((ext_vector_type(8)));

__device__ __forceinline__ float fast_tanh(float x) {
  // tanh(x) = (e^{2x}-1)/(e^{2x}+1), clamped to avoid inf/inf
  float xc = fminf(fmaxf(x, -15.0f), 15.0f);
  float e = __expf(2.0f * xc);
  return (e - 1.0f) / (e + 1.0f);
}

// ---------------------------------------------------------------------------
// K0: embedding gather. grid = B*S blocks, E_ threads.
// ---------------------------------------------------------------------------
__global__ void gather_emb_kernel(const int* __restrict__ tokens,
                                  const float* __restrict__ char_emb,
                                  float* __restrict__ emb) {
  const int bs = blockIdx.x;
  const int e  = threadIdx.x;
  const int tok = tokens[bs];                 // tokens is (B,S,1)
  emb[(size_t)bs * E_ + e] = char_emb[(size_t)tok * E_ + e];
}

// ---------------------------------------------------------------------------
// Generic fp32 WMMA GEMM:  C[M,N] = A[M,K] @ Wm[K,N] (+bias1[n]) (+bias2[n])
// One wave per 16x16 output tile, 8 waves / block.
// Uses V_WMMA_F32_16X16X4_F32 (exact fp32 — matches reference numerics).
// ---------------------------------------------------------------------------
__global__ __launch_bounds__(256) void gemm_wmma_f32(
    const float* __restrict__ A, int lda,
    const float* __restrict__ Wm, int ldw,
    float* __restrict__ C, int M, int N, int K,
    const float* __restrict__ bias1, const float* __restrict__ bias2) {
  const int tid  = threadIdx.x;
  const int wave = tid >> 5;
  const int lane = tid & 31;
  const int half = lane >> 4;     // which half-wave (K split)
  const int l16  = lane & 15;
  const int tilesN = N >> 4;
  const int tile = blockIdx.x * 8 + wave;
  const int m0 = (tile / tilesN) << 4;
  const int n0 = (tile % tilesN) << 4;
  if (m0 >= M) return;            // wave-uniform

  v8f acc = {0.f,0.f,0.f,0.f,0.f,0.f,0.f,0.f};
  const float* arow = A + (size_t)(m0 + l16) * lda + half * 2;
  for (int k = 0; k < K; k += 4) {
    v2f a;
    a.x = arow[k];
    a.y = arow[k + 1];
    const int bk = k + half * 2;
    v2f bv;
    bv.x = Wm[(size_t)bk * ldw + n0 + l16];
    bv.y = Wm[(size_t)(bk + 1) * ldw + n0 + l16];
    acc = __builtin_amdgcn_wmma_f32_16x16x4_f32(false, a, false, bv,
                                                (short)0, acc, false, false);
  }
  const int n = n0 + l16;
  float badd = 0.0f;
  if (bias1) badd += bias1[n];
  if (bias2) badd += bias2[n];
#pragma unroll
  for (int r = 0; r < 8; ++r) {
    const int m = m0 + r + half * 8;
    C[(size_t)m * N + n] = acc[r] + badd;
  }
}

// ---------------------------------------------------------------------------
// K2: fused local (window) attention + weighted CNN combine + leaky ReLU.
// grid = B*S blocks, HD threads (one per channel h).
//   re[h]   = Q[bs,h]                               (biases folded into GEMM)
//   rc[k,h] = P[bs+k-2,h]*inrange + la_w2[E+k,h] + la_b2[h]
//   sc[k]   = sum_h tanh(re+rc)*la_v[h] + la_vb ;  w = softmax_k(sc)
//   ctx[h]  = leaky( sum_k w_k*(G_k[bs+k-2,h]*inrange + cnn_w[kF+E+k,h]) + cnn_b[h] )
// ---------------------------------------------------------------------------
__global__ __launch_bounds__(HD) void local_attn_kernel(
    const float* __restrict__ Q, const float* __restrict__ P,
    const float* __restrict__ la_w2, const float* __restrict__ la_b2,
    const float* __restrict__ la_v, const float* __restrict__ la_vb,
    const float* __restrict__ G, const float* __restrict__ cnn_w,
    const float* __restrict__ cnn_b, float* __restrict__ ctx) {
  const int h  = threadIdx.x;
  const int bs = blockIdx.x;
  const int s  = bs % S_;
  __shared__ float red[W_][HD];

  const float re  = Q[(size_t)bs * HD + h];
  const float lav = la_v[h];
  const float lb2 = la_b2[h];
#pragma unroll
  for (int k = 0; k < W_; ++k) {
    const int t = s + k - 2;
    const bool in = (t >= 0) && (t < S_);
    const float pv = in ? P[(size_t)(bs + k - 2) * HD + h] : 0.0f;
    const float rc = pv + la_w2[(size_t)(E_ + k) * HD + h] + lb2;
    red[k][h] = fast_tanh(re + rc) * lav;
  }
  __syncthreads();
  for (int st = HD / 2; st > 0; st >>= 1) {
    if (h < st) {
#pragma unroll
      for (int k = 0; k < W_; ++k) red[k][h] += red[k][h + st];
    }
    __syncthreads();
  }
  float sck[W_];
  float mx = -1e30f;
  const float vb = la_vb[0];
#pragma unroll
  for (int k = 0; k < W_; ++k) { sck[k] = red[k][0] + vb; mx = fmaxf(mx, sck[k]); }
  float sum = 0.0f;
#pragma unroll
  for (int k = 0; k < W_; ++k) { sck[k] = __expf(sck[k] - mx); sum += sck[k]; }
  const float inv = 1.0f / sum;

  float c = cnn_b[h];
#pragma unroll
  for (int k = 0; k < W_; ++k) {
    const int t = s + k - 2;
    const bool in = (t >= 0) && (t < S_);
    const float g = in ? G[(size_t)k * (B_ * S_ * HD) + (size_t)(bs + k - 2) * HD + h]
                       : 0.0f;
    c += sck[k] * inv * (g + cnn_w[(size_t)(k * F_ + E_ + k) * HD + h]);
  }
  c = (c > 0.0f) ? c : 0.01f * c;
  ctx[(size_t)bs * HD + h] = c;
}

// ---------------------------------------------------------------------------
// K4: fused global attention + output head + loss.
// grid = B*(S/16) blocks, 256 threads. Block handles a 16-token row tile.
//   sc2[i,j] = tanh(re2[i]+rc2[j]) . ga_v + ga_vb   (tiled through LDS)
//   w = softmax_j(sc2) ;  attn = w @ ctx            (fp32 WMMA)
//   logits = attn*mask @ lin_w + lin_b ;  loss += -logp[tag]/(B*S)
// ---------------------------------------------------------------------------
__global__ __launch_bounds__(256) void global_attn_kernel(
    const float* __restrict__ re2, const float* __restrict__ rc2,
    const float* __restrict__ ctx, const float* __restrict__ ga_v,
    const float* __restrict__ ga_vb, const int* __restrict__ masks,
    const int* __restrict__ tags, const float* __restrict__ lin_w,
    const float* __restrict__ lin_b, float* __restrict__ out) {
  constexpr int HP = HD + 1;   // padded to dodge LDS bank conflicts
  constexpr int SP = S_ + 1;
  __shared__ float re2s[16 * HP];
  __shared__ float rcs[16 * HP];     // later reused as the attn tile
  __shared__ float scw[16 * SP];
  __shared__ float gav[HD];
  __shared__ float logitsS[16 * T_];

  const int tid = threadIdx.x;
  const int b   = blockIdx.x >> 4;          // S/16 == 16 tiles per batch
  const int i0  = (blockIdx.x & 15) << 4;

  gav[tid] = ga_v[tid];
  const float gavb = ga_vb[0];
#pragma unroll
  for (int r = 0; r < 16; ++r)
    re2s[r * HP + tid] = re2[(size_t)(b * S_ + i0 + r) * HD + tid];
  __syncthreads();

  // ---- pairwise tanh scores, 16x16 tile at a time ----
  const int ii = tid >> 4, jj = tid & 15;
  for (int jt = 0; jt < 16; ++jt) {
    const int j0 = jt << 4;
#pragma unroll
    for (int r = 0; r < 16; ++r)
      rcs[r * HP + tid] = rc2[(size_t)(b * S_ + j0 + r) * HD + tid];
    __syncthreads();
    float acc = gavb;
    for (int hh = 0; hh < HD; ++hh)
      acc += fast_tanh(re2s[ii * HP + hh] + rcs[jj * HP + hh]) * gav[hh];
    scw[ii * SP + j0 + jj] = acc;
    __syncthreads();
  }

  // ---- row softmax over 256 j's (16 lanes per row, shfl within 16) ----
  {
    const int row = tid >> 4, ln = tid & 15;
    float mx = -1e30f;
    for (int j = ln; j < S_; j += 16) mx = fmaxf(mx, scw[row * SP + j]);
    for (int off = 8; off; off >>= 1) mx = fmaxf(mx, __shfl_xor(mx, off, 16));
    float sum = 0.0f;
    for (int j = ln; j < S_; j += 16) {
      const float e = __expf(scw[row * SP + j] - mx);
      scw[row * SP + j] = e;
      sum += e;
    }
    for (int off = 8; off; off >>= 1) sum += __shfl_xor(sum, off, 16);
    const float inv = 1.0f / sum;
    for (int j = ln; j < S_; j += 16) scw[row * SP + j] *= inv;
  }
  __syncthreads();

  // ---- attn tile = weights(16 x 256) @ ctx_b(256 x 256) via fp32 WMMA ----
  const int wave = tid >> 5, lane = tid & 31;
  const int half = lane >> 4, l16 = lane & 15;
  const float* ctxb = ctx + (size_t)b * S_ * HD;
  float* attnS = rcs;   // rcs is free now
  for (int nt = wave; nt < 16; nt += 8) {
    const int n0 = nt << 4;
    v8f acc = {0.f,0.f,0.f,0.f,0.f,0.f,0.f,0.f};
    for (int k = 0; k < S_; k += 4) {
      const int ak = k + half * 2;
      v2f a;
      a.x = scw[l16 * SP + ak];
      a.y = scw[l16 * SP + ak + 1];
      v2f bv;
      bv.x = ctxb[(size_t)ak * HD + n0 + l16];
      bv.y = ctxb[(size_t)(ak + 1) * HD + n0 + l16];
      acc = __builtin_amdgcn_wmma_f32_16x16x4_f32(false, a, false, bv,
                                                  (short)0, acc, false, false);
    }
#pragma unroll
    for (int r = 0; r < 8; ++r) {
      const int m = r + half * 8;
      const float mk = (float)masks[b * S_ + i0 + m];
      attnS[m * HP + n0 + l16] = acc[r] * mk;
    }
  }
  __syncthreads();

  // ---- logits (16 tokens x 20 classes) ----
  for (int idx = tid; idx < 16 * T_; idx += 256) {
    const int tok = idx / T_, cls = idx % T_;
    float sm = lin_b[cls];
    for (int hh = 0; hh < HD; ++hh)
      sm += attnS[tok * HP + hh] * lin_w[(size_t)hh * T_ + cls];
    logitsS[tok * T_ + cls] = sm;
  }
  __syncthreads();

  // ---- log-softmax + NLL, accumulate mean into scalar loss ----
  if (tid < 16) {
    const int tag = tags[b * S_ + i0 + tid];
    float mx = -1e30f;
    for (int c = 0; c < T_; ++c) mx = fmaxf(mx, logitsS[tid * T_ + c]);
    float sum = 0.0f;
    for (int c = 0; c < T_; ++c) sum += __expf(logitsS[tid * T_ + c] - mx);
    const float logp = logitsS[tid * T_ + tag] - mx - __logf(sum);
    atomicAdd(out, -logp / (float)(B_ * S_));
  }
}

// ---------------------------------------------------------------------------
extern "C" void kernel_launch(void* const* d_in, const int* in_sizes, int n_in,
                              void* d_out, int out_size, void* d_ws, size_t ws_size,
                              hipStream_t stream) {
  (void)in_sizes; (void)n_in; (void)out_size; (void)ws_size;
  const int*   tokens   = (const int*)d_in[0];
  const int*   masks    = (const int*)d_in[1];
  /* d_in[2] sents_len unused by the reference math */
  const int*   tags     = (const int*)d_in[3];
  const float* char_emb = (const float*)d_in[4];
  const float* la_w1    = (const float*)d_in[5];
  const float* la_b1    = (const float*)d_in[6];
  const float* la_w2    = (const float*)d_in[7];
  const float* la_b2    = (const float*)d_in[8];
  const float* la_v     = (const float*)d_in[9];
  const float* la_vb    = (const float*)d_in[10];
  const float* cnn_w    = (const float*)d_in[11];
  const float* cnn_b    = (const float*)d_in[12];
  const float* ga_w1    = (const float*)d_in[13];
  const float* ga_b1    = (const float*)d_in[14];
  const float* ga_w2    = (const float*)d_in[15];
  const float* ga_b2    = (const float*)d_in[16];
  const float* ga_v     = (const float*)d_in[17];
  const float* ga_vb    = (const float*)d_in[18];
  const float* lin_w    = (const float*)d_in[19];
  const float* lin_b    = (const float*)d_in[20];

  // Workspace layout (floats); total ~22 MB
  float* ws   = (float*)d_ws;
  float* emb  = ws;                               // B*S*E
  float* Q    = emb + (size_t)B_ * S_ * E_;       // B*S*HD  (== re, biases folded)
  float* P    = Q   + (size_t)B_ * S_ * HD;       // B*S*HD
  float* G    = P   + (size_t)B_ * S_ * HD;       // 5 * B*S*HD
  float* ctx  = G   + (size_t)5 * B_ * S_ * HD;   // B*S*HD
  float* re2  = ctx + (size_t)B_ * S_ * HD;       // B*S*HD
  float* rc2  = re2 + (size_t)B_ * S_ * HD;       // B*S*HD

  hipMemsetAsync(d_out, 0, sizeof(float), stream);

  const int M = B_ * S_;                          // 2048
  gather_emb_kernel<<<M, E_, 0, stream>>>(tokens, char_emb, emb);

  const int gemmBlocks = (M / 16) * (HD / 16) / 8;  // 256 blocks, 8 tiles each

  // re = emb @ la_w1[0:E] + la_w1[E+2] + la_b1
  gemm_wmma_f32<<<gemmBlocks, 256, 0, stream>>>(emb, E_, la_w1, HD, Q, M, HD, E_,
                                                la_w1 + (size_t)(E_ + 2) * HD, la_b1);
  // P = emb @ la_w2[0:E]
  gemm_wmma_f32<<<gemmBlocks, 256, 0, stream>>>(emb, E_, la_w2, HD, P, M, HD, E_,
                                                nullptr, nullptr);
  // G_k = emb @ cnn_w[k*F : k*F+E]
  for (int k = 0; k < W_; ++k)
    gemm_wmma_f32<<<gemmBlocks, 256, 0, stream>>>(
        emb, E_, cnn_w + (size_t)k * F_ * HD, HD,
        G + (size_t)k * M * HD, M, HD, E_, nullptr, nullptr);

  local_attn_kernel<<<M, HD, 0, stream>>>(Q, P, la_w2, la_b2, la_v, la_vb,
                                          G, cnn_w, cnn_b, ctx);

  // re2 = ctx @ ga_w1 + ga_b1 ; rc2 = ctx @ ga_w2 + ga_b2
  gemm_wmma_f32<<<gemmBlocks, 256, 0, stream>>>(ctx, HD, ga_w1, HD, re2, M, HD, HD,
                                                nullptr, ga_b1);
  gemm_wmma_f32<<<gemmBlocks, 256, 0, stream>>>(ctx, HD, ga_w2, HD, rc2, M, HD, HD,
                                                nullptr, ga_b2);

  global_attn_kernel<<<B_ * (S_ / 16), 256, 0, stream>>>(
      re2, rc2, ctx, ga_v, ga_vb, masks, tags, lin_w, lin_b, (float*)d_out);
}